// DifferentiableTransformer_12128987644031
// MI455X (gfx1250) — compile-verified
//
#include <hip/hip_runtime.h>
#include <hip/hip_bf16.h>

// Problem constants (from reference)
#define NB      2
#define NATOMS  256
#define NRAD    64
#define NPTS    (64 * 64 * 64)      // 262144 grid points per batch
#define RMAX2   9.0f                // RMAX^2
#define INV_RSTEP 20.0f             // 1 / 0.05
#define MAXIDX  (NRAD - 1)
#define NTILES  (NATOMS / 16)       // 16 atom tiles
#define TABSZ   (NATOMS * NRAD)     // 16384 floats per batch

typedef float f32x2 __attribute__((ext_vector_type(2)));
typedef float f32x8 __attribute__((ext_vector_type(8)));
typedef int   v4i   __attribute__((ext_vector_type(4)));

typedef __attribute__((address_space(1))) v4i* glob_v4i_ptr;
typedef __attribute__((address_space(3))) v4i* lds_v4i_ptr;

// ---------------------------------------------------------------------------
// Prep 1: per (batch, atom) -> v = U*c (reference's upper-triangular
// transform), |v|^2, and occ' = active ? occ : 0.
// ---------------------------------------------------------------------------
__global__ void prep_atoms(const float* __restrict__ coords,
                           const unsigned char* __restrict__ active,
                           const float* __restrict__ occ,
                           const float* __restrict__ g2c,
                           float4* __restrict__ atomv,
                           float* __restrict__ occp) {
    int i = blockIdx.x * blockDim.x + threadIdx.x;
    if (i >= NB * NATOMS) return;
    float cx = coords[i * 3 + 0];
    float cy = coords[i * 3 + 1];
    float cz = coords[i * 3 + 2];
    float g0 = g2c[0], g1 = g2c[1], g2 = g2c[2];
    float g4 = g2c[4], g5 = g2c[5], g8 = g2c[8];
    float vx = g0 * cx + g1 * cy + g2 * cz;
    float vy = g4 * cy + g5 * cz;
    float vz = g8 * cz;
    float v2 = vx * vx + vy * vy + vz * vz;
    atomv[i] = make_float4(vx, vy, vz, v2);
    occp[i] = active[i] ? occ[i] : 0.0f;
}

// ---------------------------------------------------------------------------
// Prep 2: occ-scaled radial table in workspace, so the main kernel can DMA it
// straight into LDS with GLOBAL_LOAD_ASYNC_TO_LDS (no VGPR round-trip).
// ---------------------------------------------------------------------------
__global__ void prep_tab(const float* __restrict__ rad,
                         const float* __restrict__ occp,
                         float* __restrict__ stab) {
    int i = blockIdx.x * blockDim.x + threadIdx.x;
    if (i >= NB * TABSZ) return;
    stab[i] = occp[i >> 6] * rad[i];
}

// ---------------------------------------------------------------------------
// Main: one wave32 owns 16 consecutive grid points x all 256 atoms.
// d^2 for each 16x16 (point x atom) tile via V_WMMA_F32_16X16X4_F32:
//   A[p,:] = (-2ux, -2uy, -2uz, 1)    (lanes 0-15: K=0,1 ; lanes 16-31: K=2,3)
//   B[:,a] = ( vx,   vy,   vz, |v|^2)
//   C[p,:] = |u_p|^2 broadcast across columns
// D = A*B + C = |u|^2 - 2 u.v + |v|^2 = d^2.
// Occ-scaled radial table DMA'd into LDS (64 KB / block) via async-to-LDS.
// Tiles with no pair inside RMAX (ballot over the wave) skip interpolation.
// ---------------------------------------------------------------------------
__global__ void __launch_bounds__(256)
dilate_kernel(const float* __restrict__ stab,
              const float* __restrict__ g2c,
              const float4* __restrict__ atomv,
              float* __restrict__ out) {
    __shared__ float tab[TABSZ];           // 64 KB: occ-scaled radial table

    const int batch = blockIdx.x >> 11;    // 2048 blocks per batch
    const int tid   = threadIdx.x;
    const int wave  = tid >> 5;
    const int lane  = tid & 31;

    // ---- Fill LDS table: async DMA path (CDNA5), fallback = plain copy ----
    const float* src = stab + batch * TABSZ;
#if __has_builtin(__builtin_amdgcn_global_load_async_to_lds_b128)
    {
        // Each wave copies an 8 KB slice: 16 x (32 lanes x 16 B).
#pragma unroll
        for (int it = 0; it < 16; ++it) {
            const int fidx = (wave << 11) + (it << 7) + (lane << 2); // float idx
            __builtin_amdgcn_global_load_async_to_lds_b128(
                (glob_v4i_ptr)(src + fidx),
                (lds_v4i_ptr)(&tab[fidx]),
                0, 0);
        }
#if __has_builtin(__builtin_amdgcn_s_wait_asynccnt)
        __builtin_amdgcn_s_wait_asynccnt(0);
#else
        asm volatile("s_wait_asynccnt 0x0" ::: "memory");
#endif
    }
#else
    for (int idx = tid; idx < TABSZ; idx += 256) {
        tab[idx] = src[idx];
    }
#endif
    __syncthreads();

    const int laneLo  = lane & 15;
    const bool hiHalf = lane >= 16;

    const int tileInBatch = ((blockIdx.x & 2047) << 3) + wave; // 0..16383
    const int pbase       = tileInBatch << 4;                  // 16 points/wave

    // This lane's row-point (lanes 16-31 duplicate rows 0-15 for A setup).
    const int p = pbase + laneLo;
    const float fx = (float)(p & 63);
    const float fy = (float)((p >> 6) & 63);
    const float fz = (float)(p >> 12);

    const float g0 = g2c[0], g1 = g2c[1], g2 = g2c[2];
    const float g4 = g2c[4], g5 = g2c[5], g8 = g2c[8];
    const float ux = g0 * fx + g1 * fy + g2 * fz;
    const float uy = g4 * fy + g5 * fz;
    const float uz = g8 * fz;
    const float u2 = ux * ux + uy * uy + uz * uz;

    // A matrix operand (16x4 f32, 2 VGPRs/lane).
    f32x2 A;
    if (!hiHalf) { A.x = -2.0f * ux; A.y = -2.0f * uy; }
    else         { A.x = -2.0f * uz; A.y = 1.0f;       }

    // C matrix: |u_row|^2 broadcast across columns. Row of element i is
    // i (lanes 0-15) or i+8 (lanes 16-31); lane j holds u2 of row (j&15).
    f32x8 C;
#pragma unroll
    for (int i = 0; i < 8; ++i) {
        C[i] = __shfl(u2, hiHalf ? (i + 8) : i, 32);
    }

    // Preload all 16 B operands (one b64 load each, clauses up front):
    // lane needs (vx,vy) or (vz,|v|^2) of atom t*16+laneLo.
    const f32x2* av2 = (const f32x2*)(atomv + batch * NATOMS);
    const int bsel = (laneLo << 1) + (hiHalf ? 1 : 0);
    f32x2 Bs[NTILES];
#pragma unroll
    for (int t = 0; t < NTILES; ++t) {
        Bs[t] = av2[(t << 5) + bsel];
    }

    float acc[8];
#pragma unroll
    for (int i = 0; i < 8; ++i) acc[i] = 0.0f;

#pragma unroll
    for (int t = 0; t < NTILES; ++t) {
        // D = A*B + C  (16x16 f32 of squared distances)
        f32x8 D = __builtin_amdgcn_wmma_f32_16x16x4_f32(
            false, A, false, Bs[t], (short)0, C, false, false);

        // Wave-uniform sparsity cull: most 16x16 tiles have no pair in range.
        float dmin = D[0];
#pragma unroll
        for (int i = 1; i < 8; ++i) dmin = fminf(dmin, D[i]);
        if (__ballot(dmin <= RMAX2) != 0ull) {
            const float* trow = tab + (((t << 4) + laneLo) << 6); // atom's table
#pragma unroll
            for (int i = 0; i < 8; ++i) {
                float d2v  = D[i];
                float m    = (d2v <= RMAX2) ? 1.0f : 0.0f;
                // clamp to [0, +inf) in one op; d2v is never NaN here
                float d2c  = __builtin_amdgcn_fmed3f(d2v, 0.0f, 3.402823466e38f);
                float dist = __builtin_amdgcn_sqrtf(d2c);
                float rc   = dist * INV_RSTEP;
                float fl   = floorf(rc);
                int lo = (int)fl;
                lo = lo > MAXIDX ? MAXIDX : lo;
                int hi = lo + 1;
                hi = hi > MAXIDX ? MAXIDX : hi;
                float w    = rc - fl;
                float dens = (1.0f - w) * trow[lo] + w * trow[hi];
                acc[i] += m * dens;
            }
        }
    }

    // Reduce over the 16 atom-columns (xor-shuffle within 16-lane groups).
#pragma unroll
    for (int m = 1; m < 16; m <<= 1) {
#pragma unroll
        for (int i = 0; i < 8; ++i) {
            acc[i] += __shfl_xor(acc[i], m, 16);
        }
    }

    // Lanes 0 and 16 hold full row sums: rows 0-7 and 8-15 respectively.
    if (laneLo == 0) {
        float* o = out + batch * NPTS + pbase + (hiHalf ? 8 : 0);
        float4 lo4 = make_float4(acc[0], acc[1], acc[2], acc[3]);
        float4 hi4 = make_float4(acc[4], acc[5], acc[6], acc[7]);
        *(float4*)(o + 0) = lo4;
        *(float4*)(o + 4) = hi4;
    }
}

// ---------------------------------------------------------------------------
// Launch
// ---------------------------------------------------------------------------
extern "C" void kernel_launch(void* const* d_in, const int* in_sizes, int n_in,
                              void* d_out, int out_size, void* d_ws, size_t ws_size,
                              hipStream_t stream) {
    (void)in_sizes; (void)n_in; (void)out_size; (void)ws_size;
    const float*         coords = (const float*)d_in[0];         // (B,256,3)
    const unsigned char* active = (const unsigned char*)d_in[1]; // (B,256) bool
    const float*         occ    = (const float*)d_in[2];         // (B,256)
    // d_in[3] = lmax (unused)
    const float*         rad    = (const float*)d_in[4];         // (B,256,64)
    const float*         g2c    = (const float*)d_in[5];         // (3,3)

    char* ws = (char*)d_ws;
    float4* atomv = (float4*)ws;                                  //  8 KB
    float*  occp  = (float*)(ws + NB * NATOMS * sizeof(float4));  //  2 KB
    float*  stab  = (float*)(ws + NB * NATOMS * sizeof(float4)
                                + NB * NATOMS * sizeof(float));   // 128 KB

    prep_atoms<<<2, 256, 0, stream>>>(coords, active, occ, g2c, atomv, occp);
    prep_tab<<<(NB * TABSZ + 255) / 256, 256, 0, stream>>>(rad, occp, stab);
    dilate_kernel<<<4096, 256, 0, stream>>>(stab, g2c, atomv, (float*)d_out);
}